// MoEModule_54803782697400
// MI455X (gfx1250) — compile-verified
//
#include <hip/hip_runtime.h>
#include <hip/hip_bf16.h>

typedef float v2f __attribute__((ext_vector_type(2)));
typedef float v8f __attribute__((ext_vector_type(8)));

#define E_EXP   16
#define DMODEL  768
#define HDIM    768
#define T_TOK   (32 * 197)               // 6304 tokens
#define TILE_M  32                        // 2 WMMA M-tiles: B fragments reused across both
#define MAX_TILES ((T_TOK + TILE_M - 1) / TILE_M)   // 197
#define LDS_STRIDE 772                    // 768 + 4 pad: 8/16B-aligned frags, conflict-free

// ---------------- gate: logits = x @ Wg^T + bg ; idx = argmax (first max wins) -------------
__global__ __launch_bounds__(256) void moe_gate(const float* __restrict__ x,
                                                const float* __restrict__ Wg,
                                                const float* __restrict__ bg,
                                                int* __restrict__ idx)
{
    __shared__ float xs[DMODEL];
    __shared__ float lg[E_EXP];
    const int t   = blockIdx.x;
    const int tid = threadIdx.x;
    for (int i = tid; i < DMODEL; i += 256) xs[i] = x[(size_t)t * DMODEL + i];
    __syncthreads();

    const int e = tid >> 4;
    const int j = tid & 15;
    float s = 0.f;
    const float* wg = Wg + (size_t)e * DMODEL;
    for (int d = j; d < DMODEL; d += 16) s += xs[d] * wg[d];
#pragma unroll
    for (int off = 8; off >= 1; off >>= 1) s += __shfl_down(s, off, 16);
    if (j == 0) lg[e] = s + bg[e];
    __syncthreads();

    if (tid == 0) {
        int best = 0; float bv = lg[0];
#pragma unroll
        for (int q = 1; q < E_EXP; ++q) { if (lg[q] > bv) { bv = lg[q]; best = q; } }
        idx[t] = best;
    }
}

__global__ void moe_zero_counts(int* __restrict__ counts)
{
    if (threadIdx.x < E_EXP) counts[threadIdx.x] = 0;
}

__global__ __launch_bounds__(256) void moe_scatter(const int* __restrict__ idx,
                                                   int* __restrict__ counts,
                                                   int* __restrict__ lists)
{
    const int t = blockIdx.x * 256 + threadIdx.x;
    if (t >= T_TOK) return;
    const int e   = idx[t];
    const int pos = atomicAdd(&counts[e], 1);
    lists[e * T_TOK + pos] = t;
}

// per-lane async 16B copy global -> LDS (ASYNCcnt-tracked, no VGPR round trip)
__device__ __forceinline__ void async_copy_b128(unsigned int lds_byte_addr,
                                                const float* gaddr)
{
    asm volatile("global_load_async_to_lds_b128 %0, %1, off"
                 :: "v"(lds_byte_addr), "v"(gaddr)
                 : "memory");
}

__device__ __forceinline__ void wait_asynccnt0()
{
    asm volatile("s_wait_asynccnt 0x0" ::: "memory");
}

// ---------------- expert MLP: y = gelu(At @ W1^T + b1) @ W2^T + b2, gathered 32-token tiles -
__global__ __launch_bounds__(256) void moe_expert_mlp(const float* __restrict__ x,
                                                      const float* __restrict__ W1,
                                                      const float* __restrict__ b1,
                                                      const float* __restrict__ W2,
                                                      const float* __restrict__ b2,
                                                      const int* __restrict__ counts,
                                                      const int* __restrict__ lists,
                                                      float* __restrict__ out)
{
    // single buffer: holds gathered tokens (GEMM1 A), then reused for GELU(h) (GEMM2 A)
    __shared__ float S[TILE_M * LDS_STRIDE];
    __shared__ int   tok[TILE_M];

    const int e    = blockIdx.y;
    const int n_e  = counts[e];
    const int row0 = blockIdx.x * TILE_M;
    if (row0 >= n_e) return;                    // block-uniform early exit

    const int tid = threadIdx.x;
    if (tid < TILE_M) {
        int r = row0 + tid;
        if (r >= n_e) r = n_e - 1;              // pad with duplicate row; scatter is guarded
        tok[tid] = lists[e * T_TOK + r];
    }
    __syncthreads();

    // async gather of 32 token rows into LDS: 32 * 192 = 6144 16B chunks, 24 per thread
    for (int i = tid; i < TILE_M * (DMODEL / 4); i += 256) {
        const int r  = i / (DMODEL / 4);
        const int c4 = i - r * (DMODEL / 4);
        const float* g = x + (size_t)tok[r] * DMODEL + c4 * 4;
        const unsigned int l =
            (unsigned int)(size_t)&S[r * LDS_STRIDE + c4 * 4];
        async_copy_b128(l, g);
    }
    wait_asynccnt0();
    __syncthreads();

    const int wave = tid >> 5;
    const int lane = tid & 31;
    const int col  = lane & 15;     // M (within a tile, for A) / N (for B,C,D)
    const int hi   = lane >> 4;     // K/M sub-select
    const int kb   = hi * 2;        // per-lane K offset within a K=4 step
    const int nbase = wave * 96;    // this wave covers N in [nbase, nbase+96)

    const float* W1e = W1 + (size_t)e * HDIM * DMODEL;
    const float* W2e = W2 + (size_t)e * DMODEL * HDIM;

    v8f acc[2][6];

    // ================= GEMM1: h = At(32xD) @ W1e^T(DxH) + b1 =================
#pragma unroll
    for (int nt = 0; nt < 6; ++nt) {
        const float bias = b1[e * HDIM + nbase + nt * 16 + col];
#pragma unroll
        for (int v = 0; v < 8; ++v) { acc[0][nt][v] = bias; acc[1][nt][v] = bias; }
    }
    {
        const float* a0 = &S[(col)      * LDS_STRIDE + kb];   // rows 0..15
        const float* a1 = &S[(16 + col) * LDS_STRIDE + kb];   // rows 16..31
        const float* bbase = W1e + (size_t)(nbase + col) * DMODEL + kb;
        for (int k0 = 0; k0 < DMODEL; k0 += 4) {
            const v2f af0 = *(const v2f*)(a0 + k0);
            const v2f af1 = *(const v2f*)(a1 + k0);
#pragma unroll
            for (int nt = 0; nt < 6; ++nt) {
                const v2f b = *(const v2f*)(bbase + (size_t)nt * 16 * DMODEL + k0);
                acc[0][nt] = __builtin_amdgcn_wmma_f32_16x16x4_f32(
                    false, af0, false, b, (short)0, acc[0][nt], false, false);
                acc[1][nt] = __builtin_amdgcn_wmma_f32_16x16x4_f32(
                    false, af1, false, b, (short)0, acc[1][nt], false, false);
            }
        }
    }
    __syncthreads();   // all waves finished reading token tile; S can be overwritten

    // exact GELU, write h tile into S (C/D layout: M = mt*16 + v + 8*hi, N = nbase+nt*16+col)
#pragma unroll
    for (int mt = 0; mt < 2; ++mt) {
#pragma unroll
        for (int nt = 0; nt < 6; ++nt) {
            const int n = nbase + nt * 16 + col;
#pragma unroll
            for (int v = 0; v < 8; ++v) {
                const float h = acc[mt][nt][v];
                S[(mt * 16 + v + 8 * hi) * LDS_STRIDE + n] =
                    0.5f * h * (1.0f + erff(h * 0.70710678118654752f));
            }
        }
    }
    __syncthreads();

    // ================= GEMM2: y = Ht(32xH) @ W2e^T(HxD) + b2 =================
#pragma unroll
    for (int nt = 0; nt < 6; ++nt) {
        const float bias = b2[e * DMODEL + nbase + nt * 16 + col];
#pragma unroll
        for (int v = 0; v < 8; ++v) { acc[0][nt][v] = bias; acc[1][nt][v] = bias; }
    }
    {
        const float* a0 = &S[(col)      * LDS_STRIDE + kb];
        const float* a1 = &S[(16 + col) * LDS_STRIDE + kb];
        const float* bbase = W2e + (size_t)(nbase + col) * HDIM + kb;
        for (int k0 = 0; k0 < HDIM; k0 += 4) {
            const v2f af0 = *(const v2f*)(a0 + k0);
            const v2f af1 = *(const v2f*)(a1 + k0);
#pragma unroll
            for (int nt = 0; nt < 6; ++nt) {
                const v2f b = *(const v2f*)(bbase + (size_t)nt * 16 * HDIM + k0);
                acc[0][nt] = __builtin_amdgcn_wmma_f32_16x16x4_f32(
                    false, af0, false, b, (short)0, acc[0][nt], false, false);
                acc[1][nt] = __builtin_amdgcn_wmma_f32_16x16x4_f32(
                    false, af1, false, b, (short)0, acc[1][nt], false, false);
            }
        }
    }

    // scatter results back to token order (guard padded rows)
#pragma unroll
    for (int mt = 0; mt < 2; ++mt) {
#pragma unroll
        for (int nt = 0; nt < 6; ++nt) {
            const int n = nbase + nt * 16 + col;
#pragma unroll
            for (int v = 0; v < 8; ++v) {
                const int m = mt * 16 + v + 8 * hi;
                if (row0 + m < n_e)
                    out[(size_t)tok[m] * DMODEL + n] = acc[mt][nt][v];
            }
        }
    }
}

// ------------------------------------------------------------------------------------------
extern "C" void kernel_launch(void* const* d_in, const int* in_sizes, int n_in,
                              void* d_out, int out_size, void* d_ws, size_t ws_size,
                              hipStream_t stream)
{
    const float* x  = (const float*)d_in[0];   // [B,N,D]
    const float* W1 = (const float*)d_in[1];   // [E,H,D]
    const float* b1 = (const float*)d_in[2];   // [E,H]
    const float* W2 = (const float*)d_in[3];   // [E,D,H]
    const float* b2 = (const float*)d_in[4];   // [E,D]
    const float* Wg = (const float*)d_in[5];   // [E,D]
    const float* bg = (const float*)d_in[6];   // [E]
    float* out = (float*)d_out;

    char* ws = (char*)d_ws;
    int* counts = (int*)ws;                                        // 16 ints
    int* idx    = (int*)(ws + 256);                                // T ints
    int* lists  = (int*)(ws + 256 + ((T_TOK * 4 + 255) / 256) * 256); // E*T ints

    moe_zero_counts<<<1, 64, 0, stream>>>(counts);
    moe_gate<<<T_TOK, 256, 0, stream>>>(x, Wg, bg, idx);
    moe_scatter<<<(T_TOK + 255) / 256, 256, 0, stream>>>(idx, counts, lists);

    dim3 grid(MAX_TILES, E_EXP);
    moe_expert_mlp<<<grid, 256, 0, stream>>>(x, W1, b1, W2, b2, counts, lists, out);
}